// CCNN_8383776161850
// MI455X (gfx1250) — compile-verified
//
#include <hip/hip_runtime.h>

// CCNN continuous-conv on MI455X (gfx1250).
// - All matmuls via V_WMMA_F32_16X16X4_F32 (full f32 precision, matches ref).
// - Per wave: one 16-row l-tile; out[16x32] lives in two f32 C-fragments across
//   all 5 lags + skip. Big per-lag GEMM is [16x544]@[544x32] with
//   U[m][j*32+c] = h3[m][j]*f_src[m][c] built on the fly from registers/LDS
//   (rows 512..543 are h3==1 bias rows folding b4 in).
// - Weights staged to LDS with GLOBAL_LOAD_ASYNC_TO_LDS_B128 (ASYNCcnt).

#define BS      16
#define L_SEQ   2048
#define CH      32
#define NLAG    5
#define NW      4          // waves per block
#define BLOCK   (NW * 32)

typedef float v2f __attribute__((ext_vector_type(2)));
typedef float v8f __attribute__((ext_vector_type(8)));

__device__ __forceinline__ float leaky(float x) { return x >= 0.0f ? x : 0.1f * x; }
__device__ __forceinline__ v2f leaky2(v2f x) {
  v2f r; r.x = leaky(x.x); r.y = leaky(x.y); return r;
}
__device__ __forceinline__ v8f vzero8() {
  v8f z;
#pragma unroll
  for (int i = 0; i < 8; ++i) z[i] = 0.0f;
  return z;
}

// D = A(16x4) * B(4x16) + C   (f32)
__device__ __forceinline__ v8f wmma4(v2f a, v2f b, v8f c) {
  return __builtin_amdgcn_wmma_f32_16x16x4_f32(false, a, false, b, (short)0, c,
                                               false, false);
}

// Async global->LDS copy, 16B per lane, tracked by ASYNCcnt.
__device__ __forceinline__ void async_g2l_b128(unsigned lds_byte_off, const void* g) {
  asm volatile("global_load_async_to_lds_b128 %0, %1, off"
               :: "v"(lds_byte_off), "v"(g) : "memory");
}
__device__ __forceinline__ void wait_asynccnt0() {
  asm volatile("s_wait_asynccnt 0x0" ::: "memory");
}

// ---- dynamic-LDS float offsets -------------------------------------------
#define OFF_WEXT 0        // 544*32 = 17408  (W4 rows 0..511, b4 rows 512..543)
#define OFF_W2   17408    // 16*32
#define OFF_W3   17920    // 32*16
#define OFF_WSK  18432    // 32*32
#define OFF_B2   19456    // 32
#define OFF_B3   19488    // 16 (+pad)
#define OFF_BSK  19520    // 32
#define OFF_W1   19552    // 16
#define OFF_B1   19568    // 16
#define OFF_WAVE 19584
#define PW       1536     // per-wave scratch stride (tsh 32 | fsh 704 | h2 512 | h3 256)
#define SM_FLOATS (OFF_WAVE + NW * PW)   // 25728 floats = 102912 bytes

__global__ __launch_bounds__(BLOCK, 1) void ccnn_layer(
    const float* __restrict__ times, const int* __restrict__ etypes,
    const int* __restrict__ lengths, const float* __restrict__ emb,
    const float* __restrict__ W1, const float* __restrict__ b1,
    const float* __restrict__ W2, const float* __restrict__ b2,
    const float* __restrict__ W3, const float* __restrict__ b3,
    const float* __restrict__ W4, const float* __restrict__ b4,
    const float* __restrict__ Wsk, const float* __restrict__ bsk,
    const float* __restrict__ featin,   // nullptr for layer 0 (use emb[etypes])
    float* __restrict__ out,
    int lagStart, int hasSkip)
{
  extern __shared__ float sm[];
  const int tid  = threadIdx.x;
  const int lane = tid & 31;
  const int wave = tid >> 5;
  const int bidx = blockIdx.y;
  const int l0   = (blockIdx.x * NW + wave) * 16;
  const int len  = lengths[bidx];

  float* Wext = sm + OFF_WEXT;
  float* W2s  = sm + OFF_W2;
  float* W3s  = sm + OFF_W3;
  float* Wsks = sm + OFF_WSK;
  float* b2s  = sm + OFF_B2;
  float* b3s  = sm + OFF_B3;
  float* bsks = sm + OFF_BSK;
  float* W1s  = sm + OFF_W1;
  float* b1s  = sm + OFF_B1;
  float* wb   = sm + OFF_WAVE + wave * PW;
  float* tsh  = wb;            // 21 time samples (l0-5 .. l0+15)
  float* fsh  = wb + 32;       // 21*32 source features
  float* h2s  = wb + 736;      // 16*32
  float* h3s  = wb + 1248;     // 16*16

  // ---- Phase 0a: async weight staging (uniform EXEC, issued before any
  //      divergence; 16B/lane, all offsets 16B-aligned) --------------------
  {
    const unsigned lb = __builtin_amdgcn_groupstaticsize();  // dynamic-LDS base
    int idx = tid * 4;
#pragma unroll 4
    for (int it = 0; it < 32; ++it, idx += BLOCK * 4)        // W4: 16384 f32
      async_g2l_b128(lb + (OFF_WEXT + idx) * 4u, W4 + idx);
    idx = tid * 4;
#pragma unroll
    for (int it = 0; it < 2; ++it, idx += BLOCK * 4)         // b4: 1024 f32
      async_g2l_b128(lb + (OFF_WEXT + 16384 + idx) * 4u, b4 + idx);
    async_g2l_b128(lb + (OFF_W2 + tid * 4) * 4u, W2 + tid * 4);   // 512 f32
    async_g2l_b128(lb + (OFF_W3 + tid * 4) * 4u, W3 + tid * 4);   // 512 f32
    if (hasSkip) {                                           // uniform branch
      async_g2l_b128(lb + (OFF_WSK + tid * 4) * 4u, Wsk + tid * 4);
      async_g2l_b128(lb + (OFF_WSK + 512 + tid * 4) * 4u, Wsk + 512 + tid * 4);
    }
  }

  // ---- Phase 0b: small biases + per-wave input tiles (plain) -------------
  if (tid < 32) {
    b2s[tid]  = b2[tid];
    b3s[tid]  = (tid < 16) ? b3[tid] : 0.0f;
    bsks[tid] = hasSkip ? bsk[tid] : 0.0f;
  }
  if (tid < 16) { W1s[tid] = W1[tid]; b1s[tid] = b1[tid]; }

  if (lane < 21) {
    int gl = l0 - 5 + lane;
    tsh[lane] = (gl >= 0) ? times[bidx * L_SEQ + gl] : 0.0f;
  }
  for (int i = lane; i < 21 * CH; i += 32) {
    int r = i >> 5, c = i & 31;
    int gl = l0 - 5 + r;
    float v;
    if (featin) {
      v = (gl >= 0) ? featin[(bidx * L_SEQ + gl) * CH + c] : 0.0f;
    } else {
      int tp = (gl >= 0) ? etypes[bidx * L_SEQ + gl] : 0;  // type 0 -> zero emb row
      v = emb[tp * CH + c];
    }
    fsh[i] = v;
  }

  wait_asynccnt0();      // this wave's async LDS writes landed
  __syncthreads();       // + everyone else's

  const int m  = lane & 15;           // A-row / B-col index
  const int kb = (lane >> 4) << 1;    // K-base: 0 (lanes 0-15) / 2 (lanes 16-31)
  const int Mb = (lane >> 4) << 3;    // C-fragment row base

  // ---- lag-invariant small-weight fragments -> registers -----------------
  v2f w1f[4], b1f[4], w2f0[4], w2f1[4], w3f[8];
#pragma unroll
  for (int kk = 0; kk < 4; ++kk) {
    int j0 = kk * 4 + kb;
    w1f[kk].x  = W1s[j0];              w1f[kk].y  = W1s[j0 + 1];
    b1f[kk].x  = b1s[j0];              b1f[kk].y  = b1s[j0 + 1];
    w2f0[kk].x = W2s[j0 * 32 + m];     w2f0[kk].y = W2s[(j0 + 1) * 32 + m];
    w2f1[kk].x = W2s[j0 * 32 + 16 + m]; w2f1[kk].y = W2s[(j0 + 1) * 32 + 16 + m];
  }
#pragma unroll
  for (int kk = 0; kk < 8; ++kk) {
    int j0 = kk * 4 + kb;
    w3f[kk].x = W3s[j0 * 16 + m];      w3f[kk].y = W3s[(j0 + 1) * 16 + m];
  }
  const float b2m0 = b2s[m], b2m1 = b2s[16 + m], b3m = b3s[m];

  v8f acc0 = vzero8();                // out columns 0..15
  v8f acc1 = vzero8();                // out columns 16..31

  for (int lg = 0; lg < NLAG; ++lg) {
    const int lag = lagStart + lg;
    const float dtm = tsh[m + 5] - tsh[m + 5 - lag];

    // ---- h2 = leaky(h1 @ W2 + b2), h1 inline: [16x16]@[16x32] ----
    v8f c0 = vzero8(), c1 = vzero8();
#pragma unroll
    for (int kk = 0; kk < 4; ++kk) {
      v2f a = leaky2(dtm * w1f[kk] + b1f[kk]);
      c0 = wmma4(a, w2f0[kk], c0);
      c1 = wmma4(a, w2f1[kk], c1);
    }
#pragma unroll
    for (int r = 0; r < 8; ++r) {
      h2s[(Mb + r) * 32 + m]      = leaky(c0[r] + b2m0);
      h2s[(Mb + r) * 32 + 16 + m] = leaky(c1[r] + b2m1);
    }
    __builtin_amdgcn_wave_barrier();   // wave-private LDS: in-order DS suffices

    // ---- h3 = leaky(h2 @ W3 + b3): [16x32]@[32x16] ----
    v8f c3 = vzero8();
#pragma unroll
    for (int kk = 0; kk < 8; ++kk) {
      int j0 = kk * 4 + kb;
      v2f a; a.x = h2s[m * 32 + j0]; a.y = h2s[m * 32 + j0 + 1];
      c3 = wmma4(a, w3f[kk], c3);
    }
#pragma unroll
    for (int r = 0; r < 8; ++r)
      h3s[(Mb + r) * 16 + m] = leaky(c3[r] + b3m);
    __builtin_amdgcn_wave_barrier();

    // ---- masked source features this lane needs -> registers ----
    // fr[cc] = mask(m) * f_src[m][cc*4+kb .. +1]; reused by all j and b4 rows.
    const float mk = ((l0 + m - lag) >= 0 && (l0 + m) < len) ? 1.0f : 0.0f;
    const float* frow = &fsh[(m + 5 - lag) * 32];
    v2f fr[8];
#pragma unroll
    for (int cc = 0; cc < 8; ++cc) {
      int c0i = cc * 4 + kb;
      fr[cc].x = mk * frow[c0i];
      fr[cc].y = mk * frow[c0i + 1];
    }

    // ---- big GEMM: acc += U[16x544] @ Wext[544x32] ----
#pragma unroll 2
    for (int j = 0; j < 16; ++j) {
      const float hv = h3s[m * 16 + j];        // 1 LDS load per 8 K-steps
      const float* wj = &Wext[j * 1024];
#pragma unroll
      for (int cc = 0; cc < 8; ++cc) {
        int c0i = cc * 4 + kb;
        v2f a = hv * fr[cc];
        v2f bb0, bb1;
        bb0.x = wj[c0i * 32 + m];       bb0.y = wj[(c0i + 1) * 32 + m];
        bb1.x = wj[c0i * 32 + 16 + m];  bb1.y = wj[(c0i + 1) * 32 + 16 + m];
        acc0 = wmma4(a, bb0, acc0);
        acc1 = wmma4(a, bb1, acc1);
      }
    }
    // bias rows (h3 == 1): jc = 512..543 contract f_src with b4
    {
      const float* wj = &Wext[512 * 32];
#pragma unroll
      for (int cc = 0; cc < 8; ++cc) {
        int c0i = cc * 4 + kb;
        v2f bb0, bb1;
        bb0.x = wj[c0i * 32 + m];       bb0.y = wj[(c0i + 1) * 32 + m];
        bb1.x = wj[c0i * 32 + 16 + m];  bb1.y = wj[(c0i + 1) * 32 + 16 + m];
        acc0 = wmma4(fr[cc], bb0, acc0);
        acc1 = wmma4(fr[cc], bb1, acc1);
      }
    }
    __builtin_amdgcn_wave_barrier();
  }

  // ---- skip connection (layer 1): acc += feats[l] @ Wskip  (K=32) ----
  if (hasSkip) {                               // uniform branch, EXEC stays -1
#pragma unroll
    for (int kk = 0; kk < 8; ++kk) {
      int j0 = kk * 4 + kb;
      v2f a, bb0, bb1;
      a.x = fsh[(m + 5) * 32 + j0];
      a.y = fsh[(m + 5) * 32 + j0 + 1];
      bb0.x = Wsks[j0 * 32 + m];       bb0.y = Wsks[(j0 + 1) * 32 + m];
      bb1.x = Wsks[j0 * 32 + 16 + m];  bb1.y = Wsks[(j0 + 1) * 32 + 16 + m];
      acc0 = wmma4(a, bb0, acc0);
      acc1 = wmma4(a, bb1, acc1);
    }
  }

  // ---- epilogue: + bskip, outer leaky, store ----
#pragma unroll
  for (int r = 0; r < 8; ++r) {
    int row = l0 + Mb + r;
    float v0 = acc0[r], v1 = acc1[r];
    if (hasSkip) { v0 += bsks[m]; v1 += bsks[16 + m]; }
    out[(bidx * L_SEQ + row) * CH + m]      = leaky(v0);
    out[(bidx * L_SEQ + row) * CH + 16 + m] = leaky(v1);
  }
}

extern "C" void kernel_launch(void* const* d_in, const int* in_sizes, int n_in,
                              void* d_out, int out_size, void* d_ws, size_t ws_size,
                              hipStream_t stream) {
  const float* times   = (const float*)d_in[0];
  const int*   etypes  = (const int*)d_in[1];
  const int*   lengths = (const int*)d_in[2];
  const float* emb     = (const float*)d_in[3];

  // conv_params flattening: JAX pytree sorts dict keys (W1,W2,W3,W4,[Wskip],b1..);
  // also handle insertion order (W1,b1,W2,b2,...). Disambiguate via sizes.
  const float *W1a, *b1a, *W2a, *b2a, *W3a, *b3a, *W4a, *b4a;
  const float *W1b, *b1b, *W2b, *b2b, *W3b, *b3b, *W4b, *b4b, *Wskb, *bskb;
  bool sortedK = (in_sizes[5] == 512);  // sorted: idx5 = layer0 W2 (16*32)
  if (sortedK) {
    W1a = (const float*)d_in[4];  W2a = (const float*)d_in[5];
    W3a = (const float*)d_in[6];  W4a = (const float*)d_in[7];
    b1a = (const float*)d_in[8];  b2a = (const float*)d_in[9];
    b3a = (const float*)d_in[10]; b4a = (const float*)d_in[11];
    W1b = (const float*)d_in[12]; W2b = (const float*)d_in[13];
    W3b = (const float*)d_in[14]; W4b = (const float*)d_in[15];
    Wskb= (const float*)d_in[16];
    b1b = (const float*)d_in[17]; b2b = (const float*)d_in[18];
    b3b = (const float*)d_in[19]; b4b = (const float*)d_in[20];
    bskb= (const float*)d_in[21];
  } else {
    W1a = (const float*)d_in[4];  b1a = (const float*)d_in[5];
    W2a = (const float*)d_in[6];  b2a = (const float*)d_in[7];
    W3a = (const float*)d_in[8];  b3a = (const float*)d_in[9];
    W4a = (const float*)d_in[10]; b4a = (const float*)d_in[11];
    W1b = (const float*)d_in[12]; b1b = (const float*)d_in[13];
    W2b = (const float*)d_in[14]; b2b = (const float*)d_in[15];
    W3b = (const float*)d_in[16]; b3b = (const float*)d_in[17];
    W4b = (const float*)d_in[18]; b4b = (const float*)d_in[19];
    Wskb= (const float*)d_in[20]; bskb= (const float*)d_in[21];
  }

  float* featbuf = (float*)d_ws;              // BS*L*CH f32 = 4 MB
  const size_t shbytes = SM_FLOATS * sizeof(float);
  dim3 grid(L_SEQ / (16 * NW), BS);

  // Layer 0: lags 1..5, input = emb[event_types], no skip
  ccnn_layer<<<grid, BLOCK, shbytes, stream>>>(
      times, etypes, lengths, emb,
      W1a, b1a, W2a, b2a, W3a, b3a, W4a, b4a,
      nullptr, nullptr, nullptr, featbuf, /*lagStart=*/1, /*hasSkip=*/0);

  // Layer 1: lags 0..4, input = layer-0 features, + skip
  ccnn_layer<<<grid, BLOCK, shbytes, stream>>>(
      times, etypes, lengths, emb,
      W1b, b1b, W2b, b2b, W3b, b3b, W4b, b4b,
      Wskb, bskb, featbuf, (float*)d_out, /*lagStart=*/0, /*hasSkip=*/1);
}